// KoopmanModelCtrl_26027501814472
// MI455X (gfx1250) — compile-verified
//
#include <hip/hip_runtime.h>
#include <hip/hip_bf16.h>

#define EPS 1e-8f

typedef __attribute__((ext_vector_type(2))) float v2f;
typedef __attribute__((ext_vector_type(8))) float v8f;

// ---------------------------------------------------------------------------
// Generic fp32 WMMA GEMM:  C = alpha * op(A) * op(B) + beta * C
//   op(A): M x K  (transA ? A[k*lda+m] : A[m*lda+k])
//   op(B): K x N  (transB ? B[n*ldb+k] : B[k*ldb+n])
// Workgroup tile 128x64 (8 waves, each 32x32 via 2x2 V_WMMA_F32_16X16X4_F32).
// ---------------------------------------------------------------------------
#define BM 128
#define BN 64
#define BK 16

__global__ __launch_bounds__(256) void gemm_f32_wmma(
    float* __restrict__ C, const float* __restrict__ A, const float* __restrict__ B,
    int M, int N, int K, int lda, int ldb, int ldc,
    int transA, int transB, float alpha, float beta)
{
    __shared__ float As[BM][BK + 1];
    __shared__ float Bs[BK][BN + 1];

    const int t      = threadIdx.x;
    const int lane   = t & 31;
    const int wave   = t >> 5;        // 0..7
    const int waveM  = wave >> 1;     // 0..3  (rows of 32)
    const int waveN  = wave & 1;      // 0..1  (cols of 32)
    const int laneM  = lane & 15;     // m (A) / n (B) index within 16
    const int laneK2 = (lane >> 4) << 1;  // K pair offset: 0 or 2
    const int m0 = blockIdx.y * BM;
    const int n0 = blockIdx.x * BN;

    v8f acc[2][2] = {};

    for (int kk0 = 0; kk0 < K; kk0 += BK) {
        // Stage A tile: BM*BK = 2048 elems, 8 per thread
        #pragma unroll
        for (int i = 0; i < (BM * BK) / 256; ++i) {
            int idx = i * 256 + t;
            int m = idx >> 4, k = idx & 15;
            int gm = m0 + m, gk = kk0 + k;
            float v = 0.0f;
            if (gm < M && gk < K)
                v = transA ? A[(size_t)gk * lda + gm] : A[(size_t)gm * lda + gk];
            As[m][k] = v;
        }
        // Stage B tile: BK*BN = 1024 elems, 4 per thread
        #pragma unroll
        for (int i = 0; i < (BK * BN) / 256; ++i) {
            int idx = i * 256 + t;
            int k = idx >> 6, nn = idx & 63;
            int gk = kk0 + k, gn = n0 + nn;
            float v = 0.0f;
            if (gk < K && gn < N)
                v = transB ? B[(size_t)gn * ldb + gk] : B[(size_t)gk * ldb + gn];
            Bs[k][nn] = v;
        }
        __syncthreads();

        #pragma unroll
        for (int kk = 0; kk < BK; kk += 4) {
            v2f afrag[2], bfrag[2];
            #pragma unroll
            for (int i = 0; i < 2; ++i) {
                int mr = waveM * 32 + i * 16 + laneM;
                afrag[i].x = As[mr][kk + laneK2];
                afrag[i].y = As[mr][kk + laneK2 + 1];
            }
            #pragma unroll
            for (int j = 0; j < 2; ++j) {
                int nc = waveN * 32 + j * 16 + laneM;
                bfrag[j].x = Bs[kk + laneK2][nc];
                bfrag[j].y = Bs[kk + laneK2 + 1][nc];
            }
            #pragma unroll
            for (int i = 0; i < 2; ++i)
                #pragma unroll
                for (int j = 0; j < 2; ++j)
                    acc[i][j] = __builtin_amdgcn_wmma_f32_16x16x4_f32(
                        false, afrag[i], false, bfrag[j],
                        (short)0, acc[i][j], false, false);
        }
        __syncthreads();
    }

    // Epilogue: C VGPR p -> row p + 8*(lane>>4), col lane&15
    const int rhalf = (lane >> 4) * 8;
    #pragma unroll
    for (int i = 0; i < 2; ++i)
        #pragma unroll
        for (int j = 0; j < 2; ++j)
            #pragma unroll
            for (int p = 0; p < 8; ++p) {
                int gr = m0 + waveM * 32 + i * 16 + rhalf + p;
                int gc = n0 + waveN * 32 + j * 16 + laneM;
                if (gr < M && gc < N) {
                    float v = alpha * acc[i][j][p];
                    if (beta != 0.0f) v += beta * C[(size_t)gr * ldc + gc];
                    C[(size_t)gr * ldc + gc] = v;
                }
            }
}

// ---------------------------------------------------------------------------
// LU (no pivoting; E has PD symmetric part) diagonal 64x64 factor, in place.
// ---------------------------------------------------------------------------
__global__ __launch_bounds__(64) void lu_diag_kernel(float* __restrict__ D, int ld)
{
    __shared__ float T[64][65];
    const int t = threadIdx.x;
    for (int j = 0; j < 64; ++j) T[t][j] = D[(size_t)t * ld + j];
    __syncthreads();
    for (int k = 0; k < 63; ++k) {
        if (t > k) {
            float l = T[t][k] / T[k][k];
            T[t][k] = l;
            for (int j = k + 1; j < 64; ++j) T[t][j] -= l * T[k][j];
        }
        __syncthreads();
    }
    for (int j = 0; j < 64; ++j) D[(size_t)t * ld + j] = T[t][j];
}

// Solve (unit-lower Lkk) * X = X in place; one block per 64-column chunk.
__global__ __launch_bounds__(64) void trsm_ll_unit_kernel(
    const float* __restrict__ Lkk, int ldl, float* __restrict__ X, int ldx)
{
    __shared__ float Ls[64][65];
    __shared__ float T[64][65];
    const int t = threadIdx.x;
    float* Xw = X + (size_t)blockIdx.x * 64;
    for (int j = 0; j < 64; ++j) Ls[t][j] = Lkk[(size_t)t * ldl + j];
    for (int j = 0; j < 64; ++j) T[t][j]  = Xw[(size_t)t * ldx + j];
    __syncthreads();
    // thread t owns column t (independent forward substitution)
    for (int k = 0; k < 63; ++k) {
        float v = T[k][t];
        for (int j = k + 1; j < 64; ++j) T[j][t] -= Ls[j][k] * v;
    }
    __syncthreads();
    for (int j = 0; j < 64; ++j) Xw[(size_t)t * ldx + j] = T[t][j];
}

// Solve X * Ukk = X in place (X rows); one block per 64-row chunk.
__global__ __launch_bounds__(64) void trsm_ru_kernel(
    const float* __restrict__ Ukk, int ldu, float* __restrict__ X, int ldx)
{
    __shared__ float Us[64][65];
    __shared__ float T[64][65];
    const int t = threadIdx.x;
    float* Xw = X + (size_t)blockIdx.x * 64 * ldx;
    for (int j = 0; j < 64; ++j) Us[t][j] = Ukk[(size_t)t * ldu + j];
    for (int j = 0; j < 64; ++j) T[t][j]  = Xw[(size_t)t * ldx + j];
    __syncthreads();
    // thread t owns row t
    for (int k = 0; k < 64; ++k) {
        float v = T[t][k] / Us[k][k];
        T[t][k] = v;
        for (int j = k + 1; j < 64; ++j) T[t][j] -= v * Us[k][j];
    }
    __syncthreads();
    for (int j = 0; j < 64; ++j) Xw[(size_t)t * ldx + j] = T[t][j];
}

// Solve Ukk * X = X in place (backward); one block per 64-column chunk.
__global__ __launch_bounds__(64) void trsm_ul_kernel(
    const float* __restrict__ Ukk, int ldu, float* __restrict__ X, int ldx)
{
    __shared__ float Us[64][65];
    __shared__ float T[64][65];
    const int t = threadIdx.x;
    float* Xw = X + (size_t)blockIdx.x * 64;
    for (int j = 0; j < 64; ++j) Us[t][j] = Ukk[(size_t)t * ldu + j];
    for (int j = 0; j < 64; ++j) T[t][j]  = Xw[(size_t)t * ldx + j];
    __syncthreads();
    for (int k = 63; k >= 0; --k) {
        float v = T[k][t];
        for (int j = k + 1; j < 64; ++j) v -= Us[k][j] * T[j][t];
        T[k][t] = v / Us[k][k];
    }
    __syncthreads();
    for (int j = 0; j < 64; ++j) Xw[(size_t)t * ldx + j] = T[t][j];
}

// ---------------------------------------------------------------------------
// Elementwise helpers
// ---------------------------------------------------------------------------
__global__ void build_E_kernel(float* __restrict__ E, const float* __restrict__ P,
                               const float* __restrict__ R, int n)
{
    int idx = blockIdx.x * 256 + threadIdx.x;
    if (idx >= n * n) return;
    int i = idx / n, j = idx - i * n;
    float e = 0.5f * (E[idx] + P[idx]) + 0.5f * (R[(size_t)i * n + j] - R[(size_t)j * n + i]);
    if (i == j) e += 0.5f * EPS;   // P already carries its eps; this is M11's half
    E[idx] = e;
}

__global__ void add_diag_kernel(float* __restrict__ P, int n, float eps)
{
    int i = blockIdx.x * 256 + threadIdx.x;
    if (i < n) P[(size_t)i * n + i] += eps;
}

__global__ void copy_kernel(float* __restrict__ dst, const float* __restrict__ src, int count)
{
    int idx = blockIdx.x * 256 + threadIdx.x;
    if (idx < count) dst[idx] = src[idx];
}

// ---------------------------------------------------------------------------
// Host orchestration
// ---------------------------------------------------------------------------
extern "C" void kernel_launch(void* const* d_in, const int* in_sizes, int n_in,
                              void* d_out, int out_size, void* d_ws, size_t ws_size,
                              hipStream_t stream)
{
    const float* L  = (const float*)d_in[0];  // (2n, 2n)
    const float* R  = (const float*)d_in[1];  // (n, n)
    const float* Bm = (const float*)d_in[2];  // (n, 256)

    const int n  = in_sizes[2] / 256;   // dim_K
    const int K2 = 2 * n;
    const int V  = 256;
    const int NB = 64;

    float* Fw = (float*)d_ws;                     // n*n : F -> Acl in place
    float* Pw = Fw + (size_t)n * n;               // n*n : P
    float* Ew = Pw + (size_t)n * n;               // n*n : E -> LU in place
    float* Ww = Ew + (size_t)n * n;               // n*V : P @ B
    float* Gw = Ww + (size_t)n * V;               // V*n : W^T @ Acl
    float* Aout = (float*)d_out;

    const float* L1 = L;
    const float* L2 = L + (size_t)n * K2;

    auto gemm = [&](float* C, const float* A, const float* B,
                    int M, int N, int K, int lda, int ldb, int ldc,
                    int tA, int tB, float alpha, float beta) {
        dim3 grid((N + BN - 1) / BN, (M + BM - 1) / BM);
        gemm_f32_wmma<<<grid, 256, 0, stream>>>(C, A, B, M, N, K,
                                                lda, ldb, ldc, tA, tB, alpha, beta);
    };

    // --- Gram blocks of M = L L^T ---
    gemm(Fw, L2, L1, n, n, K2, K2, K2, n, 0, 1, 1.0f, 0.0f);   // F   = L2 L1^T
    gemm(Pw, L2, L2, n, n, K2, K2, K2, n, 0, 1, 1.0f, 0.0f);   // P   = L2 L2^T
    gemm(Ew, L1, L1, n, n, K2, K2, K2, n, 0, 1, 1.0f, 0.0f);   // M11 = L1 L1^T
    add_diag_kernel<<<(n + 255) / 256, 256, 0, stream>>>(Pw, n, EPS);
    build_E_kernel<<<(n * n + 255) / 256, 256, 0, stream>>>(Ew, Pw, R, n);

    // --- Blocked LU of E (no pivoting: symmetric part of E is PD) ---
    for (int k0 = 0; k0 < n; k0 += NB) {
        float* Dkk = Ew + (size_t)k0 * n + k0;
        lu_diag_kernel<<<1, 64, 0, stream>>>(Dkk, n);
        int rem = n - k0 - NB;
        if (rem > 0) {
            // U row panel: Lkk^{-1} * E[k0, k0+NB:]
            trsm_ll_unit_kernel<<<rem / NB, 64, 0, stream>>>(
                Dkk, n, Ew + (size_t)k0 * n + (k0 + NB), n);
            // L col panel: E[k0+NB:, k0] * Ukk^{-1}
            trsm_ru_kernel<<<rem / NB, 64, 0, stream>>>(
                Dkk, n, Ew + (size_t)(k0 + NB) * n + k0, n);
            // Trailing update: E22 -= Lpanel @ Upanel
            gemm(Ew + (size_t)(k0 + NB) * n + (k0 + NB),
                 Ew + (size_t)(k0 + NB) * n + k0,
                 Ew + (size_t)k0 * n + (k0 + NB),
                 rem, rem, NB, n, n, n, 0, 0, -1.0f, 1.0f);
        }
    }

    // --- Forward solve: L Y = F (Y overwrites F) ---
    for (int k0 = 0; k0 < n; k0 += NB) {
        const float* Dkk = Ew + (size_t)k0 * n + k0;
        trsm_ll_unit_kernel<<<n / NB, 64, 0, stream>>>(Dkk, n, Fw + (size_t)k0 * n, n);
        int rem = n - k0 - NB;
        if (rem > 0)
            gemm(Fw + (size_t)(k0 + NB) * n,
                 Ew + (size_t)(k0 + NB) * n + k0,
                 Fw + (size_t)k0 * n,
                 rem, n, NB, n, n, n, 0, 0, -1.0f, 1.0f);
    }

    // --- Backward solve: U X = Y (Acl overwrites F) ---
    for (int k0 = n - NB; k0 >= 0; k0 -= NB) {
        const float* Dkk = Ew + (size_t)k0 * n + k0;
        trsm_ul_kernel<<<n / NB, 64, 0, stream>>>(Dkk, n, Fw + (size_t)k0 * n, n);
        if (k0 > 0)
            gemm(Fw,
                 Ew + k0,                       // E[0:k0, k0:k0+NB]
                 Fw + (size_t)k0 * n,
                 k0, n, NB, n, n, n, 0, 0, -1.0f, 1.0f);
    }

    // --- A = Acl - B (B^T P Acl);  B^T P = (P B)^T since P symmetric ---
    gemm(Ww, Pw, Bm, n, V, n, n, V, V, 0, 0, 1.0f, 0.0f);       // W = P B        (n x V)
    gemm(Gw, Ww, Fw, V, n, n, V, n, n, 1, 0, 1.0f, 0.0f);       // G = W^T Acl    (V x n)
    copy_kernel<<<(n * n + 255) / 256, 256, 0, stream>>>(Aout, Fw, n * n);
    gemm(Aout, Bm, Gw, n, n, V, V, n, n, 0, 0, -1.0f, 1.0f);    // A = Acl - B G
}